// RBFN_25950192402607
// MI455X (gfx1250) — compile-verified
//
#include <hip/hip_runtime.h>
#include <hip/hip_bf16.h>
#include <math.h>

// ---------------------------------------------------------------------------
// Fused RBFN forward for MI455X (gfx1250, wave32):
//   out[i,j] = b[j] + sum_k x[i,k]*W[j,k]
//            + sum_m exp(-10*(||x_i||^2+||c_m||^2-2 x_i.c_m)) * W[j,64+m]
// - distance GEMM on V_WMMA_F32_16X16X4_F32 (full fp32 before the exp)
// - centers staged via GLOBAL_LOAD_ASYNC_TO_LDS_B128, double-buffered LDS
// - B fragments double-buffered in VGPRs so WMMA never waits on dscnt 0
// - the [n,M] RBF matrix is never materialized
// ---------------------------------------------------------------------------

typedef __attribute__((ext_vector_type(2))) float v2f;
typedef __attribute__((ext_vector_type(8))) float v8f;

#define D_DIM            64                // feature dim (fixed by problem)
#define KCHUNKS          (D_DIM / 4)       // 16 WMMA K-steps of 4
#define ROWS_PER_WAVE    16
#define WAVES_PER_BLOCK  8
#define ROWS_PER_BLOCK   (ROWS_PER_WAVE * WAVES_PER_BLOCK)  // 128
#define CCHUNK           64                // centers per LDS chunk (16 KB)
#define TILES_PER_CHUNK  (CCHUNK / 16)     // 4
#define NOUT             3
#define BETA             10.0f
#define MAX_M            1024
#define CHUNK_BYTES      (CCHUNK * D_DIM * 4)

// memory -> LDS, no VGPR staging, tracked by ASYNCcnt
__device__ __forceinline__ void async_chunk_to_lds(const float* gsrc,
                                                   float* ldsDst, int tid) {
    uint32_t l = (uint32_t)(uintptr_t)ldsDst + (uint32_t)(tid * 16);
    uint64_t g = (uint64_t)(uintptr_t)gsrc + (uint64_t)(tid * 16);
#pragma unroll
    for (int i = 0; i < CHUNK_BYTES / (256 * 16); ++i) {  // 4 x b128 / thread
        asm volatile("global_load_async_to_lds_b128 %0, %1, off"
                     :: "v"(l + (uint32_t)(i * 256 * 16)),
                        "v"(g + (uint64_t)(i * 256 * 16))
                     : "memory");
    }
}

__device__ __forceinline__ void wait_async_all() {
    asm volatile("s_wait_asynccnt 0x0" ::: "memory");
}

__global__ __launch_bounds__(256)
void rbfn_fused_kernel(const float* __restrict__ x,
                       const float* __restrict__ C,
                       const float* __restrict__ W,
                       const float* __restrict__ bias,
                       float* __restrict__ out,
                       int n, int M)
{
    __shared__ float sC[2][CCHUNK * D_DIM];          // 2 x 16 KB center chunks
    __shared__ float sW[NOUT * (D_DIM + MAX_M)];     // ~13 KB full W

    const int tid  = threadIdx.x;
    const int lane = tid & 31;
    const int wave = tid >> 5;
    const int half = lane >> 4;          // 0: lanes 0-15, 1: lanes 16-31
    const int l16  = lane & 15;
    const int koff = half ? 2 : 0;       // f32 A/B fragment K offset (ISA layout)
    const int WL   = D_DIM + M;          // 1088

    // ---- stage W into LDS + kick off chunk 0 async copy ------------------
    async_chunk_to_lds(C, &sC[0][0], tid);
    const int wtot = NOUT * WL;
    for (int i = tid; i < wtot; i += blockDim.x) sW[i] = W[i];
    wait_async_all();
    __syncthreads();

    // ---- per-wave A fragments (16 rows x 64) + row norms -----------------
    const int rowBase = blockIdx.x * ROWS_PER_BLOCK + wave * ROWS_PER_WAVE;
    const float* xr   = x + (size_t)(rowBase + l16) * D_DIM;

    v2f   afrag[KCHUNKS];
    float x2p = 0.f;
#pragma unroll
    for (int kk = 0; kk < KCHUNKS; ++kk) {
        float e0 = xr[4 * kk + koff];
        float e1 = xr[4 * kk + koff + 1];
        v2f t; t.x = e0; t.y = e1;
        afrag[kk] = t;
        x2p = fmaf(e0, e0, fmaf(e1, e1, x2p));
    }
    float x2full = x2p + __shfl_xor(x2p, 16, 32);    // ||x_row||^2, row = l16

    float x2v[8];                                     // norms for my C rows
#pragma unroll
    for (int v = 0; v < 8; ++v)
        x2v[v] = __shfl(x2full, v + 8 * half, 32);

    float o0[8] = {}, o1[8] = {}, o2[8] = {};

    // fold in x @ W_x^T (k split across the 16 column-lanes; reduced later)
#pragma unroll
    for (int v = 0; v < 8; ++v) {
        const float* xm = x + (size_t)(rowBase + v + 8 * half) * D_DIM;
#pragma unroll
        for (int kq = 0; kq < 4; ++kq) {
            int   k  = l16 + 16 * kq;
            float xv = xm[k];
            o0[v] = fmaf(xv, sW[0 * WL + k], o0[v]);
            o1[v] = fmaf(xv, sW[1 * WL + k], o1[v]);
            o2[v] = fmaf(xv, sW[2 * WL + k], o2[v]);
        }
    }

    // ---- main loop over center chunks (double-buffered LDS) --------------
    const int nChunks = M / CCHUNK;
    for (int ch = 0; ch < nChunks; ++ch) {
        const int    cur  = ch & 1;
        const float* base = &sC[cur][0];

        if (ch + 1 < nChunks)   // overlap next chunk's copy with this compute
            async_chunk_to_lds(C + (size_t)(ch + 1) * CCHUNK * D_DIM,
                               &sC[cur ^ 1][0], tid);

        auto load_bfrag = [&](int t, v2f* bf) {
            const float* cr = base + (t * 16 + l16) * D_DIM;
#pragma unroll
            for (int kk = 0; kk < KCHUNKS; ++kk) {
                v2f tb; tb.x = cr[4 * kk + koff]; tb.y = cr[4 * kk + koff + 1];
                bf[kk] = tb;
            }
        };

        auto process_tile = [&](int t, v2f* bf) {
            float c2p = 0.f;
#pragma unroll
            for (int kk = 0; kk < KCHUNKS; ++kk)
                c2p = fmaf(bf[kk].x, bf[kk].x, fmaf(bf[kk].y, bf[kk].y, c2p));
            float c2 = c2p + __shfl_xor(c2p, 16, 32);   // ||c_col||^2

            v8f acc = {};
#pragma unroll
            for (int kk = 0; kk < KCHUNKS; ++kk)
                acc = __builtin_amdgcn_wmma_f32_16x16x4_f32(
                          false, afrag[kk], false, bf[kk],
                          (short)0, acc, false, false);

            const int col = ch * CCHUNK + t * 16 + l16;
            const float w0 = sW[0 * WL + D_DIM + col];
            const float w1 = sW[1 * WL + D_DIM + col];
            const float w2 = sW[2 * WL + D_DIM + col];
#pragma unroll
            for (int v = 0; v < 8; ++v) {
                float d2 = fmaf(-2.0f, acc[v], x2v[v] + c2);
                float r  = __expf(-BETA * d2);
                o0[v] = fmaf(r, w0, o0[v]);
                o1[v] = fmaf(r, w1, o1[v]);
                o2[v] = fmaf(r, w2, o2[v]);
            }
        };

        // register double-buffering: issue tile t+1's ds loads before
        // consuming tile t, so WMMA never waits on dscnt==0
        v2f bfA[KCHUNKS], bfB[KCHUNKS];
        load_bfrag(0, bfA);
#pragma unroll
        for (int tp = 0; tp < TILES_PER_CHUNK / 2; ++tp) {
            const int t0 = 2 * tp, t1 = 2 * tp + 1;
            load_bfrag(t1, bfB);
            process_tile(t0, bfA);
            if (t1 + 1 < TILES_PER_CHUNK) load_bfrag(t1 + 1, bfA);
            process_tile(t1, bfB);
        }

        wait_async_all();      // my async loads into the other buffer landed
        __syncthreads();       // everyone done reading cur / writing nxt
    }

    // ---- reduce across the 16 column-lanes (xor tree stays within halves)
#pragma unroll
    for (int m = 8; m >= 1; m >>= 1) {
#pragma unroll
        for (int v = 0; v < 8; ++v) {
            o0[v] += __shfl_xor(o0[v], m, 32);
            o1[v] += __shfl_xor(o1[v], m, 32);
            o2[v] += __shfl_xor(o2[v], m, 32);
        }
    }

    if (l16 == 0) {  // lane 0 -> rows 0-7, lane 16 -> rows 8-15
        const float b0 = bias[0], b1 = bias[1], b2 = bias[2];
#pragma unroll
        for (int v = 0; v < 8; ++v) {
            const int row = rowBase + v + 8 * half;
            out[row * NOUT + 0] = o0[v] + b0;
            out[row * NOUT + 1] = o1[v] + b1;
            out[row * NOUT + 2] = o2[v] + b2;
        }
    }
}

extern "C" void kernel_launch(void* const* d_in, const int* in_sizes, int n_in,
                              void* d_out, int out_size, void* d_ws, size_t ws_size,
                              hipStream_t stream) {
    const float* x = (const float*)d_in[0];
    const float* C = (const float*)d_in[1];
    const float* W = (const float*)d_in[2];
    const float* b = (const float*)d_in[3];
    float* out     = (float*)d_out;

    const int n = in_sizes[0] / D_DIM;   // 65536
    const int M = in_sizes[1] / D_DIM;   // 1024

    const int blocks = n / ROWS_PER_BLOCK;   // 512
    rbfn_fused_kernel<<<blocks, 256, 0, stream>>>(x, C, W, b, out, n, M);
}